// HawkesIntensityFunction_39685497815277
// MI455X (gfx1250) — compile-verified
//
#include <hip/hip_runtime.h>

// Problem constants from the reference: B=2, N=4096, D_MODEL=256
#define BB 2
#define NN 4096
#define DM 256

typedef __attribute__((ext_vector_type(2))) float v2f;
typedef __attribute__((ext_vector_type(8))) float v8f;

#define LOG2E 1.4426950408889634f

// ---------------------------------------------------------------------------
// Kernel A: pairwise excitation sum via V_WMMA_F32_16X16X4_F32.
//   out[b,i] = alpha * ( (i+1) + sum_{j>i} exp(-beta*(t_i-t_j) - gamma*dist_ij) )
// One wave owns 16 i-rows; 8 waves/block split the j-tile range.
// Diagonal j-tile (jt==it) is peeled (wave 0, masked); all other tiles have
// j > i for every pair -> unconditional accumulate, no per-element cmp.
// exp/sqrt use raw v_exp_f32 / v_sqrt_f32 (log2e folded into beta/gamma).
//
// A (16x4 f32): lanes 0-15 -> K=0 (x_i), K=1 (y_i); lanes 16-31 -> K=2,3 = 0.
// B (4x16 f32): vgpr0 lanes0-15 = row K=0 (x_j), lanes16-31 = row K=2 (0);
//               vgpr1 lanes0-15 = row K=1 (y_j), lanes16-31 = row K=3 (0).
// D (16x16 f32, 8 vgprs): vgpr v, lane L -> M = v + 8*(L>>4), N = L&15.
// ---------------------------------------------------------------------------
__global__ __launch_bounds__(256) void hawkes_excitation_kernel(
    const float* __restrict__ t,     // [B,N]
    const float* __restrict__ loc,   // [B,N,2]
    const float* __restrict__ alpha_p,
    const float* __restrict__ beta_p,
    const float* __restrict__ gamma_p,
    float* __restrict__ out)         // [B,N]
{
    const int nIT  = NN / 16;
    const int it   = blockIdx.x % nIT;
    const int b    = blockIdx.x / nIT;
    const int lane = threadIdx.x & 31;
    const int wave = threadIdx.x >> 5;   // 0..7
    const int half = lane >> 4;          // 0: lanes 0-15, 1: lanes 16-31
    const int n16  = lane & 15;

    const float alpha  = *alpha_p;
    const float beta2  = (*beta_p)  * LOG2E;   // pre-scale: exp(x)=exp2(x*log2e)
    const float gamma2 = (*gamma_p) * LOG2E;

    const float* tb = t   + (size_t)b * NN;
    const float* lb = loc + (size_t)b * NN * 2;

    const int i0 = it * 16;

    // A operand: loop-invariant (depends only on the i-tile).
    v2f a;
    if (half == 0) {
        a.x = lb[2 * (i0 + n16) + 0];
        a.y = lb[2 * (i0 + n16) + 1];
    } else {
        a.x = 0.0f; a.y = 0.0f;
    }

    // Loop-invariant per-lane row data for m = v + 8*half.
    float nbti8[8], sxi8[8];
    #pragma unroll
    for (int v = 0; v < 8; ++v) {
        const int i = i0 + v + 8 * half;
        const float xi = lb[2 * i + 0];
        const float yi = lb[2 * i + 1];
        nbti8[v] = -beta2 * tb[i];            // -beta'*t_i (precomputed)
        sxi8[v]  = xi * xi + yi * yi;
    }

    float acc[8];
    #pragma unroll
    for (int v = 0; v < 8; ++v) acc[v] = 0.0f;

    // ---- Peeled diagonal tile (jt == it): needs the strict-triu mask. ----
    if (wave == 0) {
        const int jn = i0 + n16;
        const float xj  = lb[2 * jn + 0];
        const float yj  = lb[2 * jn + 1];
        const float btj = beta2 * tb[jn];
        const float sxj = xj * xj + yj * yj;

        v2f bm;
        if (half == 0) { bm.x = xj; bm.y = yj; }
        else           { bm.x = 0.0f; bm.y = 0.0f; }

        v8f c = {};
        v8f d = __builtin_amdgcn_wmma_f32_16x16x4_f32(
            false, a, false, bm, (short)0, c, false, false);

        #pragma unroll
        for (int v = 0; v < 8; ++v) {
            const int m    = v + 8 * half;
            const float d2   = sxi8[v] + sxj - 2.0f * d[v];
            const float dist = __builtin_amdgcn_sqrtf(fmaxf(d2, 0.0f));
            const float arg  = fmaf(-gamma2, dist, nbti8[v] + btj);
            const float e    = __builtin_amdgcn_exp2f(arg);
            acc[v] += (n16 > m) ? e : 0.0f;   // strict upper triangle
        }
    }

    // ---- Off-diagonal tiles (jt > it): every j > every i, no masking. ----
    const int nJT = NN / 16;
    for (int jt = it + 1 + wave; jt < nJT; jt += 8) {
        const int jn = jt * 16 + n16;
        const float xj  = lb[2 * jn + 0];
        const float yj  = lb[2 * jn + 1];
        const float btj = beta2 * tb[jn];
        const float sxj = xj * xj + yj * yj;

        v2f bm;
        if (half == 0) { bm.x = xj; bm.y = yj; }
        else           { bm.x = 0.0f; bm.y = 0.0f; }

        v8f c = {};
        // D[m][n] = x_i[m]*x_j[n] + y_i[m]*y_j[n]
        v8f d = __builtin_amdgcn_wmma_f32_16x16x4_f32(
            false, a, false, bm, (short)0, c, false, false);

        #pragma unroll
        for (int v = 0; v < 8; ++v) {
            const float d2   = sxi8[v] + sxj - 2.0f * d[v];
            const float dist = __builtin_amdgcn_sqrtf(fmaxf(d2, 0.0f));
            const float arg  = fmaf(-gamma2, dist, nbti8[v] + btj);
            acc[v] += __builtin_amdgcn_exp2f(arg);
        }
    }

    // Reduce over N (16 lanes within each half-wave); wave32 -> masks 1,2,4,8.
    #pragma unroll
    for (int v = 0; v < 8; ++v) {
        float s = acc[v];
        s += __shfl_xor(s, 1, 32);
        s += __shfl_xor(s, 2, 32);
        s += __shfl_xor(s, 4, 32);
        s += __shfl_xor(s, 8, 32);
        acc[v] = s;
    }

    __shared__ float red[8][16];
    if (n16 == 0) {
        // lane 0 holds sums for m=0..7, lane 16 for m=8..15
        #pragma unroll
        for (int v = 0; v < 8; ++v) red[wave][v + 8 * half] = acc[v];
    }
    __syncthreads();

    if (threadIdx.x < 16) {
        const int m = threadIdx.x;
        float s = 0.0f;
        #pragma unroll
        for (int w = 0; w < 8; ++w) s += red[w][m];
        const int i = i0 + m;
        // lower triangle + diagonal collapse to alpha each: (i+1) terms
        out[(size_t)b * NN + i] = alpha * ((float)(i + 1) + s);
    }
}

// ---------------------------------------------------------------------------
// Kernel B: out[b,i] += mu + exp(z[b,i,:]·w + bias). One wave per row,
// each lane covers 8 of the 256 features with float4 loads. Bandwidth-bound.
// ---------------------------------------------------------------------------
__global__ __launch_bounds__(256) void hawkes_base_kernel(
    const float* __restrict__ z,     // [B,N,DM]
    const float* __restrict__ w,     // [DM]
    const float* __restrict__ bias_p,
    const float* __restrict__ mu_p,
    float* __restrict__ out)         // [B,N]
{
    const int lane = threadIdx.x & 31;
    const int wave = threadIdx.x >> 5;
    const int row  = blockIdx.x * 8 + wave;      // 0 .. B*N-1

    const float4* zr = (const float4*)(z + (size_t)row * DM);
    const float4* wr = (const float4*)w;

    float dot = 0.0f;
    #pragma unroll
    for (int cchunk = 0; cchunk < 2; ++cchunk) {
        const float4 zv = zr[lane * 2 + cchunk];
        const float4 wv = wr[lane * 2 + cchunk];
        dot = fmaf(zv.x, wv.x, dot);
        dot = fmaf(zv.y, wv.y, dot);
        dot = fmaf(zv.z, wv.z, dot);
        dot = fmaf(zv.w, wv.w, dot);
    }
    dot += __shfl_xor(dot, 1, 32);
    dot += __shfl_xor(dot, 2, 32);
    dot += __shfl_xor(dot, 4, 32);
    dot += __shfl_xor(dot, 8, 32);
    dot += __shfl_xor(dot, 16, 32);

    if (lane == 0) {
        out[row] += *mu_p + __builtin_amdgcn_exp2f((dot + *bias_p) * LOG2E);
    }
}

// ---------------------------------------------------------------------------
// Inputs (setup_inputs order): z, event_times, event_locations, w, b,
//                              mu, alpha, beta, gamma
// ---------------------------------------------------------------------------
extern "C" void kernel_launch(void* const* d_in, const int* in_sizes, int n_in,
                              void* d_out, int out_size, void* d_ws, size_t ws_size,
                              hipStream_t stream) {
    const float* z     = (const float*)d_in[0];
    const float* t     = (const float*)d_in[1];
    const float* loc   = (const float*)d_in[2];
    const float* w     = (const float*)d_in[3];
    const float* bias  = (const float*)d_in[4];
    const float* mu    = (const float*)d_in[5];
    const float* alpha = (const float*)d_in[6];
    const float* beta  = (const float*)d_in[7];
    const float* gamma = (const float*)d_in[8];
    float* out = (float*)d_out;

    // Kernel A fully overwrites out, Kernel B accumulates on top (same stream).
    hawkes_excitation_kernel<<<dim3(BB * (NN / 16)), 256, 0, stream>>>(
        t, loc, alpha, beta, gamma, out);
    hawkes_base_kernel<<<dim3((BB * NN) / 8), 256, 0, stream>>>(
        z, w, bias, mu, out);
}